// NormalizationFlow_33526514713162
// MI455X (gfx1250) — compile-verified
//
#include <hip/hip_runtime.h>
#include <math.h>

typedef float v2f __attribute__((ext_vector_type(2)));
typedef float v4f __attribute__((ext_vector_type(4)));
typedef float v8f __attribute__((ext_vector_type(8)));

#define NUM_LAYERS 15
#define TPW 4            // 16-sample tiles per wave (independent WMMA chains)
#define THREADS 256
#define BLOCKS 1024
#define SLOPE 0.01f
#define VEC_F 112                        // floats per layer in packed vector block
#define WMID_F (NUM_LAYERS * 3 * 256)    // 11520 floats
#define VECB_F (NUM_LAYERS * VEC_F)      // 1680 floats
#define TOT_F  (WMID_F + VECB_F)         // 13200 floats (52.8 KB)

__device__ __forceinline__ float leaky(float x) { return fmaxf(x, SLOPE * x); }

// ---------------- prep kernel: build staged weight image in d_ws ----------------
// ws[0 .. WMID_F):  w_mid image, [i][j][M][q], block-diagonal, column-permuted for j>=1
// ws[WMID_F .. ):   per-layer vector blocks (see main kernel comment)
__global__ __launch_bounds__(THREADS) void flow_prep_kernel(
    const int*   __restrict__ perms,
    const float* __restrict__ w_in,
    const float* __restrict__ b_in,
    const float* __restrict__ w_mid,
    const float* __restrict__ b_mid,
    const float* __restrict__ w_out,
    const float* __restrict__ b_out,
    float* __restrict__ ws)
{
  int stride = gridDim.x * blockDim.x;
  int tid0 = blockIdx.x * blockDim.x + threadIdx.x;

  for (int idx = tid0; idx < WMID_F; idx += stride) {
    int i = idx / 768, r = idx % 768, j = r / 256, e = r % 256;
    int M = e >> 4, q = e & 15;
    // column permutation for mid layers that consume a C-layout activation
    int K = (j == 0) ? q : (((q >> 2) << 1) + (q & 1) + (((q >> 1) & 1) << 3));
    float v = 0.0f;
    if (M < 8 && K < 8)        v = w_mid[(((i * 2 + 0) * 3 + j) * 8 + M) * 8 + K];
    else if (M >= 8 && K >= 8) v = w_mid[(((i * 2 + 1) * 3 + j) * 8 + (M - 8)) * 8 + (K - 8)];
    ws[idx] = v;
  }
  for (int idx = tid0; idx < VECB_F; idx += stride) {
    int i = idx / VEC_F, p = idx % VEC_F;
    float v = 0.0f;
    if (p < 16) {                       // w_in, identity order
      v = w_in[(i * 2 + (p >> 3)) * 8 + (p & 7)];
    } else if (p < 32) {                // b_in, identity order
      int f = p - 16;
      v = b_in[(i * 2 + (f >> 3)) * 8 + (f & 7)];
    } else if (p < 80) {                // b_mid, half-interleaved: slot 4m+2h+e <- feat 8h+2m+e
      int r = p - 32, j = r >> 4, s = r & 15;
      int m = s >> 2, h = (s >> 1) & 1, e = s & 1;
      v = b_mid[((i * 2 + h) * 3 + j) * 8 + (2 * m + e)];
    } else if (p < 96) {                // w_out, half-interleaved
      int s = p - 80;
      int m = s >> 2, h = (s >> 1) & 1, e = s & 1;
      v = w_out[(i * 2 + h) * 8 + (2 * m + e)];
    } else if (p < 100) {               // b_out replicated: b0,b1,b0,b1
      v = b_out[i * 2 + ((p - 96) & 1)];
    } else if (p < 104) {               // swap flag replicated (float 0/1)
      v = (i < NUM_LAYERS - 1 && perms[i * 2] != 0) ? 1.0f : 0.0f;
    }
    ws[WMID_F + idx] = v;
  }
}

// ---------------- main kernel ----------------
__global__ __launch_bounds__(THREADS) void flow_wmma_kernel(
    const float* __restrict__ z,
    const float* __restrict__ ws,
    float* __restrict__ out,
    int batch)
{
  // s_all[0..WMID_F):   s_wmid[i][j][M][q] — A-operand weights
  // s_all[WMID_F..):    per-layer vector block, ONE lane base (2*half floats):
  //   [0..15] w_in  [16..31] b_in  [32..79] b_mid(j, half-interleaved)
  //   [80..95] w_out(half-interleaved)  [96..99] b_out rep  [100..103] swap rep
  __shared__ __align__(16) float s_all[TOT_F];

  const int tid = threadIdx.x;

  // branch-free bulk copy of the staged image (L2-resident across blocks)
  {
    const v4f* src = (const v4f*)ws;
    v4f* dst = (v4f*)s_all;
    for (int idx = tid; idx < TOT_F / 4; idx += THREADS) dst[idx] = src[idx];
  }
  __syncthreads();

  const int lane = tid & 31;
  const int wave = tid >> 5;
  const int half = lane >> 4;
  const int l16  = lane & 15;
  const int wavesPerBlock = blockDim.x >> 5;
  const int nWaves = gridDim.x * wavesPerBlock;
  const int gWave  = blockIdx.x * wavesPerBlock + wave;
  const int ntiles = batch >> 4;

  for (int tbase = gWave * TPW; tbase < ntiles; tbase += nWaves * TPW) {
    float zl[TPW], zr[TPW];
    #pragma unroll
    for (int t = 0; t < TPW; ++t) {
      int tile = tbase + t;               // wave-uniform validity -> EXEC stays full
      if (tile < ntiles) {
        v2f zz = *(const v2f*)(z + 2 * (tile * 16 + l16));  // replicated across halves
        zl[t] = zz.x; zr[t] = zz.y;
      } else { zl[t] = 0.f; zr[t] = 0.f; }
    }

    // loop-invariant lane bases; advance by constant stride per layer
    const float* wrow = s_all + (l16 * 16 + 2 * half);
    const float* vrow = s_all + WMID_F + 2 * half;

    for (int i = 0; i < NUM_LAYERS; ++i) {
      // ---- input layer (K=1): pair hp[t][k] holds features (4k+2h, 4k+1+2h) ----
      v2f hp[TPW][4];
      #pragma unroll
      for (int k = 0; k < 4; ++k) {
        v2f w  = *(const v2f*)(vrow + 4 * k);        // w_in pair
        v2f bv = *(const v2f*)(vrow + 16 + 4 * k);   // b_in pair
        #pragma unroll
        for (int t = 0; t < TPW; ++t) {
          hp[t][k].x = leaky(fmaf(zl[t], w.x, bv.x));
          hp[t][k].y = leaky(fmaf(zl[t], w.y, bv.y));
        }
      }

      // ---- 3 mid layers: C = W * H + bias; B operands are the activation pairs ----
      #pragma unroll
      for (int j = 0; j < 3; ++j) {
        v2f A[4];  // A layout: M = l16, K = 4k + v + 2*half
        #pragma unroll
        for (int k = 0; k < 4; ++k)
          A[k] = *(const v2f*)(wrow + j * 256 + 4 * k);
        v2f bp0 = *(const v2f*)(vrow + 32 + 16 * j);       // features (8h, 8h+1)
        v2f bp1 = *(const v2f*)(vrow + 32 + 16 * j + 4);
        v2f bp2 = *(const v2f*)(vrow + 32 + 16 * j + 8);
        v2f bp3 = *(const v2f*)(vrow + 32 + 16 * j + 12);
        #pragma unroll
        for (int t = 0; t < TPW; ++t) {
          v8f c = { bp0.x, bp0.y, bp1.x, bp1.y, bp2.x, bp2.y, bp3.x, bp3.y };
          #pragma unroll
          for (int k = 0; k < 4; ++k)
            c = __builtin_amdgcn_wmma_f32_16x16x4_f32(
                    false, A[k], false, hp[t][k], (short)0, c, false, false);
          // leaky + repack C (feature = v+8*half) into consecutive pairs
          #pragma unroll
          for (int k = 0; k < 4; ++k) {
            hp[t][k].x = leaky(c[2 * k]);
            hp[t][k].y = leaky(c[2 * k + 1]);
          }
        }
      }

      // ---- output layer: per-lane dot; half0 -> log_s dot, half1 -> b dot ----
      v2f wp0 = *(const v2f*)(vrow + 80);
      v2f wp1 = *(const v2f*)(vrow + 80 + 4);
      v2f wp2 = *(const v2f*)(vrow + 80 + 8);
      v2f wp3 = *(const v2f*)(vrow + 80 + 12);
      v2f bo  = *(const v2f*)(vrow + 96);     // (b_out0, b_out1) in both halves
      float swf = *(vrow + 100);              // swap flag, replicated
      bool sw = swf > 0.5f;
      #pragma unroll
      for (int t = 0; t < TPW; ++t) {
        float dot = hp[t][0].x * wp0.x + hp[t][0].y * wp0.y
                  + hp[t][1].x * wp1.x + hp[t][1].y * wp1.y
                  + hp[t][2].x * wp2.x + hp[t][2].y * wp2.y
                  + hp[t][3].x * wp3.x + hp[t][3].y * wp3.y;
        float other = __shfl_xor(dot, 16, 32);
        float logs = (half ? other : dot) + bo.x;
        float bb   = (half ? dot : other) + bo.y;
        float zrn  = __expf(logs) * zr[t] + bb;   // affine coupling
        float nzl  = sw ? zrn : zl[t];            // column permutation of z
        float nzr  = sw ? zl[t] : zrn;
        zl[t] = nzl; zr[t] = nzr;                 // replicated in both halves
      }

      wrow += 3 * 256;   // next layer's weight matrices
      vrow += VEC_F;     // next layer's vector block
    }

    #pragma unroll
    for (int t = 0; t < TPW; ++t) {
      int tile = tbase + t;
      if (tile < ntiles && half == 0) {
        v2f zz; zz.x = zl[t]; zz.y = zr[t];
        *(v2f*)(out + 2 * (tile * 16 + l16)) = zz;
      }
    }
  }
}

extern "C" void kernel_launch(void* const* d_in, const int* in_sizes, int n_in,
                              void* d_out, int out_size, void* d_ws, size_t ws_size,
                              hipStream_t stream) {
  const float* z      = (const float*)d_in[0];
  const int*   perms  = (const int*)  d_in[1];
  const float* w_in   = (const float*)d_in[2];
  const float* b_in   = (const float*)d_in[3];
  const float* w_mid  = (const float*)d_in[4];
  const float* b_mid  = (const float*)d_in[5];
  const float* w_out  = (const float*)d_in[6];
  const float* b_out  = (const float*)d_in[7];
  float* out = (float*)d_out;
  float* ws  = (float*)d_ws;
  int batch = in_sizes[0] / 2;

  flow_prep_kernel<<<64, THREADS, 0, stream>>>(
      perms, w_in, b_in, w_mid, b_mid, w_out, b_out, ws);
  flow_wmma_kernel<<<BLOCKS, THREADS, 0, stream>>>(z, ws, out, batch);
}